// GlobalDistNet_30915174596976
// MI455X (gfx1250) — compile-verified
//
#include <hip/hip_runtime.h>
#include <hip/hip_bf16.h>

// ---------------------------------------------------------------------------
// GlobalDistNet GCN pipeline for MI455X (gfx1250, wave32).
// fp32 WMMA (v_wmma_f32_16x16x4_f32) for the node-feature GEMMs; L2-resident
// edge gather/scatter with fp32 global atomics for the graph convolution.
// K-templated GEMM: float4 A staging, pair-packed B staging (one b64 per
// WMMA B-fragment), software-pipelined fragment loads.
// ---------------------------------------------------------------------------

#define N_NODES   38332
#define F_DIM     544
#define C_DIM     128
#define N_EDGES   (N_NODES * 16)
#define N_SEG     (N_EDGES + N_NODES)     // edges + self loops
#define OUT_LEN   38333
#define NEG_SLOPE 0.01f

typedef float v2f __attribute__((ext_vector_type(2)));
typedef float v4f __attribute__((ext_vector_type(4)));
typedef float v8f __attribute__((ext_vector_type(8)));

#if defined(__has_builtin)
#if __has_builtin(__builtin_amdgcn_wmma_f32_16x16x4_f32)
#define HAVE_WMMA_F32_16X16X4 1
#endif
#endif

__device__ __forceinline__ float lrelu(float v) {
    return v >= 0.f ? v : NEG_SLOPE * v;
}

// --- masked_select / masked_scatter recovery -------------------------------
__global__ void k_poi(const float* __restrict__ feature,
                      const unsigned char* __restrict__ mask,
                      int* __restrict__ idxcol, int* __restrict__ poi) {
    int i = blockIdx.x * blockDim.x + threadIdx.x;
    if (i >= N_NODES) return;
    const unsigned char* mrow = mask + (size_t)i * F_DIM;
    int col = 0;
    for (int j = 0; j < F_DIM; ++j) { if (mrow[j]) { col = j; break; } }
    idxcol[i] = col;
    poi[i] = (int)feature[(size_t)i * F_DIM + col];
}

// src_vals[i] = emb[poi[i/128]][i%128]  (flattened masked_scatter source)
__global__ void k_val0(const float* __restrict__ emb,
                       const int* __restrict__ poi,
                       float* __restrict__ val0) {
    int i = blockIdx.x * blockDim.x + threadIdx.x;
    if (i >= N_NODES) return;
    val0[i] = emb[(size_t)poi[i >> 7] * C_DIM + (i & 127)];
}

// --- symmetric GCN normalization -------------------------------------------
__global__ void k_deg_init(float* __restrict__ deg) {
    int i = blockIdx.x * blockDim.x + threadIdx.x;
    if (i < N_NODES) deg[i] = 1.0f;            // self loop
}
__global__ void k_deg_acc(const int* __restrict__ dstp, float* __restrict__ deg) {
    int e = blockIdx.x * blockDim.x + threadIdx.x;
    if (e < N_EDGES) atomicAdd(&deg[dstp[e]], 1.0f);
}
__global__ void k_dinv(const float* __restrict__ deg, float* __restrict__ dinv) {
    int i = blockIdx.x * blockDim.x + threadIdx.x;
    if (i < N_NODES) dinv[i] = (deg[i] > 0.f) ? rsqrtf(deg[i]) : 0.f;
}

// --- bias broadcast / fills -------------------------------------------------
__global__ void k_bias_init(float* __restrict__ t, const float* __restrict__ b) {
    long long idx = (long long)blockIdx.x * blockDim.x + threadIdx.x;
    if (idx < (long long)N_NODES * C_DIM) t[idx] = b[idx & 127];
}
__global__ void k_fill_b0(float* __restrict__ ts, const float* __restrict__ b0) {
    int i = blockIdx.x * blockDim.x + threadIdx.x;
    if (i < N_NODES) ts[i] = b0[0];
}

// --- fp32 WMMA GEMM:  H[N x 128] = A[N x K] @ W[K x 128] -------------------
// Block = 256 threads = 8 waves; block tile 128 rows x 128 cols; K chunk 32.
// Wave w owns rows [w*16, w*16+16) and loops the 8 column tiles.
//
// LDS layouts (64 banks x 4B):
//  As[r][kk]  stride 36 dw: lanes 0-15 -> banks {36r+ka, +1}, lanes 16-31 ->
//             {36r+ka+2, +3}; all distinct (36r mod 64 spans 16 values, step 4).
//  Bs2[p][c*2+(k&1)] pairs (B[2p][c], B[2p+1][c]) contiguously -> one b64 per
//             fragment. Row stride 288 dw == 32 (mod 64): half-waves reading
//             rows p / p+1 land on disjoint 32-bank sets. Conflict-free.
#define KC      32
#define BSTRIDE 288

template <int K, bool SUBST>
__global__ __launch_bounds__(256)
void k_gemm_wmma(const float* __restrict__ A,
                 const float* __restrict__ W,
                 const int* __restrict__ idxcol,   // used when SUBST
                 const float* __restrict__ val0,
                 float* __restrict__ Hout) {
    __shared__ __align__(16) float As[128][KC + 4];
    __shared__ __align__(16) float Bs2[KC / 2][BSTRIDE];

    const int tid  = threadIdx.x;
    const int wv   = tid >> 5;
    const int lane = tid & 31;
    const int m0   = blockIdx.x * 128;

    v8f acc[8] = {};

    #pragma unroll 1
    for (int k0 = 0; k0 < K; k0 += KC) {
        // A tile 128 x 32 as float4 (8 consecutive lanes cover one row: coalesced)
        #pragma unroll
        for (int j = 0; j < 4; ++j) {
            int idx = tid + j * 256;              // 1024 float4 slots
            int r   = idx >> 3;                   // row in tile
            int k4  = (idx & 7) << 2;             // k offset (0,4,...,28)
            int grow = m0 + r;
            v4f v = {0.f, 0.f, 0.f, 0.f};
            if (grow < N_NODES) {
                v = *reinterpret_cast<const v4f*>(&A[(size_t)grow * K + k0 + k4]);
                if (SUBST) {                      // emb_feature substitution
                    int ic = idxcol[grow];
                    #pragma unroll
                    for (int q = 0; q < 4; ++q)
                        if (k0 + k4 + q == ic) v[q] = val0[grow];
                }
            }
            *reinterpret_cast<v4f*>(&As[r][k4]) = v;
        }
        // B tile 32 x 128 -> K-paired layout: thread loads (W[2p][c], W[2p+1][c])
        // (both coalesced across lanes) and writes one conflict-free ds_store_b64.
        #pragma unroll
        for (int j = 0; j < 8; ++j) {
            int idx = tid + j * 256;              // 2048 (p,c) pairs
            int p   = idx >> 7;                   // k-pair index 0..15
            int c   = idx & 127;
            v2f v;
            v.x = W[(size_t)(k0 + 2 * p) * C_DIM + c];
            v.y = W[(size_t)(k0 + 2 * p + 1) * C_DIM + c];
            *reinterpret_cast<v2f*>(&Bs2[p][c << 1]) = v;
        }
        __syncthreads();

#ifdef HAVE_WMMA_F32_16X16X4
        // software-pipelined fragment loads: fetch kk+4 while issuing kk's WMMAs
        const int cb   = lane & 15;
        const int arow = wv * 16 + cb;
        const int khi  = (lane >> 4) << 1;        // 0 (lanes 0-15) / 2 (lanes 16-31)
        v2f a_cur = *reinterpret_cast<const v2f*>(&As[arow][khi]);
        v2f b_cur[8];
        #pragma unroll
        for (int nc = 0; nc < 8; ++nc)
            b_cur[nc] = *reinterpret_cast<const v2f*>(
                &Bs2[khi >> 1][(nc * 16 + cb) << 1]);

        #pragma unroll
        for (int kk = 0; kk < KC; kk += 4) {
            v2f a_nxt = a_cur;
            v2f b_nxt[8];
            #pragma unroll
            for (int nc = 0; nc < 8; ++nc) b_nxt[nc] = b_cur[nc];
            if (kk + 4 < KC) {
                const int ka = kk + 4 + khi;
                a_nxt = *reinterpret_cast<const v2f*>(&As[arow][ka]);
                #pragma unroll
                for (int nc = 0; nc < 8; ++nc)
                    b_nxt[nc] = *reinterpret_cast<const v2f*>(
                        &Bs2[ka >> 1][(nc * 16 + cb) << 1]);
            }
            #pragma unroll
            for (int nc = 0; nc < 8; ++nc)
                acc[nc] = __builtin_amdgcn_wmma_f32_16x16x4_f32(
                    false, a_cur, false, b_cur[nc], (short)0, acc[nc],
                    false, false);
            a_cur = a_nxt;
            #pragma unroll
            for (int nc = 0; nc < 8; ++nc) b_cur[nc] = b_nxt[nc];
        }
#else
        // scalar fallback consistent with the C/D fragment layout
        const int rr = (lane >> 4) << 3;
        const int cb = lane & 15;
        for (int kk = 0; kk < KC; kk += 4) {
            for (int nc = 0; nc < 8; ++nc) {
                #pragma unroll
                for (int v = 0; v < 8; ++v) {
                    float sum = 0.f;
                    #pragma unroll
                    for (int q = 0; q < 4; ++q)
                        sum += As[wv * 16 + rr + v][kk + q] *
                               Bs2[(kk + q) >> 1][((nc * 16 + cb) << 1) | ((kk + q) & 1)];
                    acc[nc][v] += sum;
                }
            }
        }
#endif
        __syncthreads();
    }

    // C/D layout: VGPR v -> row v (lanes 0-15) / row v+8 (lanes 16-31)
    const int rbase = m0 + wv * 16 + ((lane >> 4) << 3);
    const int cbs   = lane & 15;
    #pragma unroll
    for (int nc = 0; nc < 8; ++nc) {
        #pragma unroll
        for (int v = 0; v < 8; ++v) {
            int row = rbase + v;
            if (row < N_NODES)
                Hout[(size_t)row * C_DIM + nc * 16 + cbs] = acc[nc][v];
        }
    }
}

// --- edge scatter: T[dst,:] += H[src,:] * dinv[src]*dinv[dst] ---------------
__global__ void k_scatter(const float* __restrict__ H,
                          const int* __restrict__ srcp, const int* __restrict__ dstp,
                          const float* __restrict__ dinv, float* __restrict__ T) {
    long long idx = (long long)blockIdx.x * blockDim.x + threadIdx.x;
    if (idx >= (long long)N_SEG * C_DIM) return;
    int e = (int)(idx >> 7);
    int c = (int)(idx & 127);
    int s, d;
    if (e < N_EDGES) { s = srcp[e]; d = dstp[e]; }
    else             { s = d = e - N_EDGES; }
    float nrm = dinv[s] * dinv[d];
    atomicAdd(&T[(size_t)d * C_DIM + c], H[(size_t)s * C_DIM + c] * nrm);
}

// --- activations ------------------------------------------------------------
__global__ void k_lrelu_vec(const float* __restrict__ t, float* __restrict__ x,
                            long long n) {
    long long idx = (long long)blockIdx.x * blockDim.x + threadIdx.x;
    if (idx < n) x[idx] = lrelu(t[idx]);
}
__global__ void k_unit_act(const float* __restrict__ t, float* __restrict__ x,
                           long long n) {            // x = lrelu(t) + t
    long long idx = (long long)blockIdx.x * blockDim.x + threadIdx.x;
    if (idx < n) { float v = t[idx]; x[idx] = lrelu(v) + v; }
}

// --- S[i] = dot(X[i,:], W_out) : one wave per row ---------------------------
__global__ __launch_bounds__(256)
void k_gemv_out(const float* __restrict__ X, const float* __restrict__ Wout,
                float* __restrict__ S) {
    const int wv = threadIdx.x >> 5, lane = threadIdx.x & 31;
    const int row = blockIdx.x * 8 + wv;
    if (row >= N_NODES) return;                       // wave-uniform exit
    float acc = 0.f;
    #pragma unroll
    for (int j = 0; j < 4; ++j) {
        int c = lane + j * 32;
        acc += X[(size_t)row * C_DIM + c] * Wout[c];
    }
    #pragma unroll
    for (int off = 16; off >= 1; off >>= 1)
        acc += __shfl_xor(acc, off, 32);
    if (lane == 0) S[row] = acc;
}

__global__ void k_scatter_s(const float* __restrict__ S,
                            const int* __restrict__ srcp, const int* __restrict__ dstp,
                            const float* __restrict__ dinv, float* __restrict__ TS) {
    int e = blockIdx.x * blockDim.x + threadIdx.x;
    if (e >= N_SEG) return;
    int s, d;
    if (e < N_EDGES) { s = srcp[e]; d = dstp[e]; }
    else             { s = d = e - N_EDGES; }
    atomicAdd(&TS[d], S[s] * dinv[s] * dinv[d]);
}

// --- FC head ----------------------------------------------------------------
__global__ void k_init128(float* __restrict__ dst, const float* __restrict__ src) {
    int i = threadIdx.x;
    if (i < 128) dst[i] = src[i];
}
__global__ __launch_bounds__(256)
void k_fc1(const float* __restrict__ XS, const float* __restrict__ W1,
           float* __restrict__ Hpre) {
    __shared__ float red[256];
    const int t = threadIdx.x;
    const int o = t & 127;
    const int half = t >> 7;
    const int RPB = (N_NODES + gridDim.x - 1) / gridDim.x;
    const int r0 = blockIdx.x * RPB;
    int r1 = r0 + RPB; if (r1 > N_NODES) r1 = N_NODES;
    float acc = 0.f;
    for (int i = r0 + half; i < r1; i += 2)           // consecutive t -> consecutive o: coalesced
        acc += XS[i] * W1[(size_t)i * 128 + o];
    red[t] = acc;
    __syncthreads();
    if (t < 128) atomicAdd(&Hpre[t], red[t] + red[t + 128]);
}
__global__ void k_relu128(const float* __restrict__ Hpre, float* __restrict__ Hv) {
    int i = threadIdx.x;
    if (i < 128) Hv[i] = fmaxf(Hpre[i], 0.f);
}
__global__ __launch_bounds__(256)
void k_fc2(const float* __restrict__ Hv, const float* __restrict__ W2,
           const float* __restrict__ b2, float* __restrict__ out) {
    __shared__ float hs[128];
    if (threadIdx.x < 128) hs[threadIdx.x] = Hv[threadIdx.x];
    __syncthreads();
    int j = blockIdx.x * blockDim.x + threadIdx.x;
    if (j >= OUT_LEN) return;
    float acc = b2[j];
    #pragma unroll 8
    for (int k = 0; k < 128; ++k)
        acc += hs[k] * W2[(size_t)k * OUT_LEN + j];
    out[j] = fmaxf(acc, 0.f);
}

// ---------------------------------------------------------------------------
extern "C" void kernel_launch(void* const* d_in, const int* in_sizes, int n_in,
                              void* d_out, int out_size, void* d_ws, size_t ws_size,
                              hipStream_t stream) {
    (void)in_sizes; (void)n_in; (void)out_size; (void)ws_size;

    const float*         feature = (const float*)d_in[0];
    const unsigned char* mask    = (const unsigned char*)d_in[1];
    const int*           edge    = (const int*)d_in[2];
    const float*         emb     = (const float*)d_in[3];
    const float*         W_in    = (const float*)d_in[4];
    const float*         b_in    = (const float*)d_in[5];
    const float*         W_gcn   = (const float*)d_in[6];
    const float*         b_gcn   = (const float*)d_in[7];
    const float*         W_out   = (const float*)d_in[8];
    const float*         b_out   = (const float*)d_in[9];
    const float*         W_fc1   = (const float*)d_in[10];
    const float*         b_fc1   = (const float*)d_in[11];
    const float*         W_fc2   = (const float*)d_in[12];
    const float*         b_fc2   = (const float*)d_in[13];
    const int* srcp = edge;
    const int* dstp = edge + N_EDGES;

    // workspace carve-out (256B aligned slabs)
    char* ws = (char*)d_ws;
    size_t off = 0;
    auto carve = [&](size_t bytes) -> void* {
        void* p = ws + off;
        off = (off + bytes + 255) & ~(size_t)255;
        return p;
    };
    int*   idxcol = (int*)  carve((size_t)N_NODES * 4);
    int*   poi    = (int*)  carve((size_t)N_NODES * 4);
    float* val0   = (float*)carve((size_t)N_NODES * 4);
    float* deg    = (float*)carve((size_t)N_NODES * 4);
    float* dinv   = (float*)carve((size_t)N_NODES * 4);
    float* h      = (float*)carve((size_t)N_NODES * C_DIM * 4);
    float* t      = (float*)carve((size_t)N_NODES * C_DIM * 4);
    float* x      = (float*)carve((size_t)N_NODES * C_DIM * 4);
    float* s      = (float*)carve((size_t)N_NODES * 4);
    float* ts     = (float*)carve((size_t)N_NODES * 4);
    float* xs     = (float*)carve((size_t)N_NODES * 4);
    float* hpre   = (float*)carve(128 * 4);
    float* hvec   = (float*)carve(128 * 4);

    const int gN    = (N_NODES + 255) / 256;
    const int gE    = (N_EDGES + 255) / 256;
    const int gSeg  = (N_SEG + 255) / 256;
    const long long nc = (long long)N_NODES * C_DIM;
    const int gNC   = (int)((nc + 255) / 256);
    const int gScat = (int)(((long long)N_SEG * C_DIM + 255) / 256);
    const int gGemm = (N_NODES + 127) / 128;

    // masked_scatter prep + normalization
    k_poi     <<<gN, 256, 0, stream>>>(feature, mask, idxcol, poi);
    k_val0    <<<gN, 256, 0, stream>>>(emb, poi, val0);
    k_deg_init<<<gN, 256, 0, stream>>>(deg);
    k_deg_acc <<<gE, 256, 0, stream>>>(dstp, deg);
    k_dinv    <<<gN, 256, 0, stream>>>(deg, dinv);

    // input layer: x = lrelu(gcn_conv(emb_feature, W_in, b_in))
    k_bias_init<<<gNC, 256, 0, stream>>>(t, b_in);
    k_gemm_wmma<F_DIM, true><<<gGemm, 256, 0, stream>>>(feature, W_in,
                                                        idxcol, val0, h);
    k_scatter  <<<gScat, 256, 0, stream>>>(h, srcp, dstp, dinv, t);
    k_lrelu_vec<<<gNC, 256, 0, stream>>>(t, x, nc);

    // 4 GcnUnits: t = gcn_conv(x, W_gcn[i]); x = lrelu(t) + t
    for (int l = 0; l < 4; ++l) {
        k_bias_init<<<gNC, 256, 0, stream>>>(t, b_gcn + (size_t)l * C_DIM);
        k_gemm_wmma<C_DIM, false><<<gGemm, 256, 0, stream>>>(
            x, W_gcn + (size_t)l * C_DIM * C_DIM, nullptr, nullptr, h);
        k_scatter  <<<gScat, 256, 0, stream>>>(h, srcp, dstp, dinv, t);
        k_unit_act <<<gNC, 256, 0, stream>>>(t, x, nc);
    }

    // output conv: xs = lrelu(scatter(x @ W_out) + b_out)
    k_fill_b0   <<<gN, 256, 0, stream>>>(ts, b_out);
    k_gemv_out  <<<(N_NODES + 7) / 8, 256, 0, stream>>>(x, W_out, s);
    k_scatter_s <<<gSeg, 256, 0, stream>>>(s, srcp, dstp, dinv, ts);
    k_lrelu_vec <<<gN, 256, 0, stream>>>(ts, xs, (long long)N_NODES);

    // FC head
    k_init128<<<1, 128, 0, stream>>>(hpre, b_fc1);
    k_fc1    <<<32, 256, 0, stream>>>(xs, W_fc1, hpre);
    k_relu128<<<1, 128, 0, stream>>>(hpre, hvec);
    k_fc2    <<<(OUT_LEN + 255) / 256, 256, 0, stream>>>(hvec, W_fc2, b_fc2,
                                                         (float*)d_out);
}